// GuidedFilter_4604204941502
// MI455X (gfx1250) — compile-verified
//
#include <hip/hip_runtime.h>

// Guided filter, MI455X (gfx1250), wave32.
// Box filters realized as GEMMs vs constant 0/1 band matrices using
// V_WMMA_F32_16X16X4_F32 (full f32 precision; kernel is bandwidth-bound so
// low-precision WMMA buys nothing). One wave per 16x16 output tile.

typedef __attribute__((ext_vector_type(2))) float v2f;
typedef __attribute__((ext_vector_type(8))) float v8f;

#define RAD    8
#define IMG_H  1024
#define IMG_W  1024
#define NPLN   24                      // 8 * 3 planes
#define TILE   16
#define REG    32                      // TILE + 2*RAD  (== WMMA K extent!)
#define TX_    (IMG_W / TILE)          // 64
#define TY_    (IMG_H / TILE)          // 64
#define TPP    (TX_ * TY_)             // 4096 tiles / plane
#define NTILES (NPLN * TPP)            // 98304
#define WPB    4                       // waves per workgroup
#define NBLK   (NTILES / WPB)          // 24576 blocks

static __device__ __forceinline__ int imin(int a, int b) { return a < b ? a : b; }
static __device__ __forceinline__ int imax(int a, int b) { return a > b ? a : b; }

// Band-window matrix entry: 1.0 iff k is inside the 17-wide window of output m.
static __device__ __forceinline__ float boxw(int m, int k) {
  return (k >= m && k <= m + 2 * RAD) ? 1.0f : 0.0f;
}

static __device__ __forceinline__ v8f wmma4(v2f a, v2f b, v8f c) {
  // D(16x16,f32) = A(16x4,f32) x B(4x16,f32) + C
  return __builtin_amdgcn_wmma_f32_16x16x4_f32(
      /*neg_a=*/false, a, /*neg_b=*/false, b,
      /*c_mod=*/(short)0, c, /*reuse_a=*/false, /*reuse_b=*/false);
}

// Wave-internal LDS ordering (DS ops of one wave retire in order; this only
// stops the compiler/scheduler from reordering across the phase boundary).
static __device__ __forceinline__ void lds_fence() {
  asm volatile("s_wait_dscnt 0" ::: "memory");
}

// Per-pixel analytic box count (boxfilter of ones with edge clamping).
static __device__ __forceinline__ float boxcount(int orow, int ocol) {
  float nv = (float)(imin(orow + RAD, IMG_H - 1) - imax(orow - RAD, 0) + 1);
  float nh = (float)(imin(ocol + RAD, IMG_W - 1) - imax(ocol - RAD, 0) + 1);
  return nv * nh;
}

// ---------------------------------------------------------------------------
// Stage 1: box-filter {x, y, x*y, x*x}; emit A = cov/(var+eps), b = my - A*mx.
// ---------------------------------------------------------------------------
__global__ __launch_bounds__(WPB * 32) void gf_stage1(
    const float* __restrict__ x, const float* __restrict__ y,
    float* __restrict__ Aout, float* __restrict__ Bout) {
  __shared__ float lx[WPB][REG * REG];
  __shared__ float ly[WPB][REG * REG];
  __shared__ float lv[WPB][TILE * REG];

  const int lane = (int)(threadIdx.x & 31u);
  const int wid  = (int)(threadIdx.x >> 5u);
  const int tile = (int)blockIdx.x * WPB + wid;
  if (tile >= NTILES) return;

  const int pln  = tile / TPP;
  const int t    = tile % TPP;
  const int row0 = (t / TX_) * TILE;
  const int col0 = (t % TX_) * TILE;
  const size_t pbase = (size_t)pln * IMG_H * IMG_W;

  float* Lx = lx[wid];
  float* Ly = ly[wid];
  float* Lv = lv[wid];

  // Cooperative load of 32x32 zero-padded region: one column per lane.
  const int gc  = col0 - RAD + lane;
  const bool ck = (gc >= 0) && (gc < IMG_W);
  for (int i = 0; i < REG; ++i) {
    const int gr  = row0 - RAD + i;
    const bool ok = ck && (gr >= 0) && (gr < IMG_H);
    const size_t g = pbase + (size_t)gr * IMG_W + gc;
    Lx[i * REG + lane] = ok ? x[g] : 0.0f;
    Ly[i * REG + lane] = ok ? y[g] : 0.0f;
  }
  lds_fence();

  const int m  = lane & 15;  // A-frag row / B-frag col / D col
  const int lh = lane >> 4;  // lane half selects K sub-pair

  v8f S[4];
  #pragma unroll
  for (int q = 0; q < 4; ++q) {  // 0:x 1:y 2:x*y 3:x*x
    // Vertical: Wv(16x32) x Q(32x32) -> V(16x32), two 16-col halves.
    v8f av[2] = {};
    #pragma unroll
    for (int kc = 0; kc < 8; ++kc) {
      const int k0 = 4 * kc + 2 * lh;
      v2f a = {boxw(m, k0), boxw(m, k0 + 1)};
      #pragma unroll
      for (int h = 0; h < 2; ++h) {
        const int cc = m + 16 * h;  // region column for this lane
        const float q0x = Lx[(k0    ) * REG + cc];
        const float q1x = Lx[(k0 + 1) * REG + cc];
        const float q0y = Ly[(k0    ) * REG + cc];
        const float q1y = Ly[(k0 + 1) * REG + cc];
        v2f b;
        b.x = (q == 0) ? q0x : (q == 1) ? q0y : (q == 2) ? q0x * q0y : q0x * q0x;
        b.y = (q == 0) ? q1x : (q == 1) ? q1y : (q == 2) ? q1x * q1y : q1x * q1x;
        av[h] = wmma4(a, b, av[h]);
      }
    }

    // Relayout V: accumulator (C layout) -> row-major LDS (A-operand feed).
    lds_fence();  // prior horizontal-pass reads of Lv must retire first
    #pragma unroll
    for (int h = 0; h < 2; ++h)
      #pragma unroll
      for (int i = 0; i < 8; ++i)
        Lv[(i + 8 * lh) * REG + m + 16 * h] = av[h][i];
    lds_fence();

    // Horizontal: V(16x32) x Wh(32x16) -> S(16x16).
    v8f acc = {};
    #pragma unroll
    for (int kc = 0; kc < 8; ++kc) {
      const int k0 = 4 * kc + 2 * lh;
      v2f a = {Lv[m * REG + k0], Lv[m * REG + k0 + 1]};
      v2f b = {boxw(m, k0), boxw(m, k0 + 1)};  // Wh[k][n] = boxw(n, k), n == m
      acc = wmma4(a, b, acc);
    }
    S[q] = acc;
  }

  // Per-pixel guided-filter coefficients.
  #pragma unroll
  for (int i = 0; i < 8; ++i) {
    const int orow = row0 + i + 8 * lh;
    const int ocol = col0 + m;
    const float N  = boxcount(orow, ocol);
    const float mx  = S[0][i] / N;
    const float my  = S[1][i] / N;
    const float cov = S[2][i] / N - mx * my;
    const float var = S[3][i] / N - mx * mx;
    const float a_  = cov / (var + 1e-8f);
    const float b_  = my - a_ * mx;
    const size_t o = pbase + (size_t)orow * IMG_W + ocol;
    Aout[o] = a_;
    Bout[o] = b_;
  }
}

// ---------------------------------------------------------------------------
// Stage 2: box-filter {A, b}; out = mean_A * x + mean_b.
// ---------------------------------------------------------------------------
__global__ __launch_bounds__(WPB * 32) void gf_stage2(
    const float* __restrict__ Ain, const float* __restrict__ Bin,
    const float* __restrict__ x, float* __restrict__ out) {
  __shared__ float la[WPB][REG * REG];
  __shared__ float lb[WPB][REG * REG];
  __shared__ float lv[WPB][TILE * REG];

  const int lane = (int)(threadIdx.x & 31u);
  const int wid  = (int)(threadIdx.x >> 5u);
  const int tile = (int)blockIdx.x * WPB + wid;
  if (tile >= NTILES) return;

  const int pln  = tile / TPP;
  const int t    = tile % TPP;
  const int row0 = (t / TX_) * TILE;
  const int col0 = (t % TX_) * TILE;
  const size_t pbase = (size_t)pln * IMG_H * IMG_W;

  float* La = la[wid];
  float* Lb = lb[wid];
  float* Lv = lv[wid];

  const int gc  = col0 - RAD + lane;
  const bool ck = (gc >= 0) && (gc < IMG_W);
  for (int i = 0; i < REG; ++i) {
    const int gr  = row0 - RAD + i;
    const bool ok = ck && (gr >= 0) && (gr < IMG_H);
    const size_t g = pbase + (size_t)gr * IMG_W + gc;
    La[i * REG + lane] = ok ? Ain[g] : 0.0f;
    Lb[i * REG + lane] = ok ? Bin[g] : 0.0f;
  }
  lds_fence();

  const int m  = lane & 15;
  const int lh = lane >> 4;

  v8f S[2];
  #pragma unroll
  for (int q = 0; q < 2; ++q) {
    const float* Lq = (q == 0) ? La : Lb;

    v8f av[2] = {};
    #pragma unroll
    for (int kc = 0; kc < 8; ++kc) {
      const int k0 = 4 * kc + 2 * lh;
      v2f a = {boxw(m, k0), boxw(m, k0 + 1)};
      #pragma unroll
      for (int h = 0; h < 2; ++h) {
        const int cc = m + 16 * h;
        v2f b = {Lq[(k0) * REG + cc], Lq[(k0 + 1) * REG + cc]};
        av[h] = wmma4(a, b, av[h]);
      }
    }

    lds_fence();
    #pragma unroll
    for (int h = 0; h < 2; ++h)
      #pragma unroll
      for (int i = 0; i < 8; ++i)
        Lv[(i + 8 * lh) * REG + m + 16 * h] = av[h][i];
    lds_fence();

    v8f acc = {};
    #pragma unroll
    for (int kc = 0; kc < 8; ++kc) {
      const int k0 = 4 * kc + 2 * lh;
      v2f a = {Lv[m * REG + k0], Lv[m * REG + k0 + 1]};
      v2f b = {boxw(m, k0), boxw(m, k0 + 1)};
      acc = wmma4(a, b, acc);
    }
    S[q] = acc;
  }

  #pragma unroll
  for (int i = 0; i < 8; ++i) {
    const int orow = row0 + i + 8 * lh;
    const int ocol = col0 + m;
    const float N = boxcount(orow, ocol);
    const size_t o = pbase + (size_t)orow * IMG_W + ocol;
    out[o] = (S[0][i] / N) * x[o] + (S[1][i] / N);
  }
}

// ---------------------------------------------------------------------------
extern "C" void kernel_launch(void* const* d_in, const int* in_sizes, int n_in,
                              void* d_out, int out_size, void* d_ws, size_t ws_size,
                              hipStream_t stream) {
  (void)in_sizes; (void)n_in; (void)out_size; (void)ws_size;
  const float* x = (const float*)d_in[0];
  const float* y = (const float*)d_in[1];
  float* out = (float*)d_out;
  // Workspace: A plane-stack then b plane-stack (2 * 96 MB); both fully
  // rewritten by stage 1 every call, and largely L2-resident for stage 2.
  float* wsA = (float*)d_ws;
  float* wsB = wsA + (size_t)NPLN * IMG_H * IMG_W;

  dim3 block(WPB * 32);
  dim3 grid(NBLK);
  gf_stage1<<<grid, block, 0, stream>>>(x, y, wsA, wsB);
  gf_stage2<<<grid, block, 0, stream>>>(wsA, wsB, x, out);
}